// ADDMeanM_16595753632500
// MI455X (gfx1250) — compile-verified
//
#include <hip/hip_runtime.h>
#include <math.h>

// CDNA5 / gfx1250: wave32, WMMA (not MFMA). We use the full-fp32 tensor op
// V_WMMA_F32_16X16X4_F32:  A = 16x4 f32 (2 VGPRs/lane), B = 4x16 f32,
// C/D = 16x16 f32 (8 VGPRs/lane).  Fragment layouts per ISA 7.12.2:
//   A: lane = M (0..15, both halves), VGPR v -> K = 2*(lane/16) + v
//   B: lane = N,                      VGPR v -> K = 2*(lane/16) + v
//   C/D: lane = N, VGPR v -> M = v + 8*(lane/16)
typedef __attribute__((ext_vector_type(2))) float v2f;
typedef __attribute__((ext_vector_type(8))) float v8f;

#define NTHREADS 512   // 16 wave32 waves -> 16 tiles of 16x16 = full 64x64 C
#define N 64
#define NN 4096

// C = A * B for 64x64 fp32 matrices resident in LDS. C must not alias A or B.
// All 512 threads participate (EXEC all ones, required by WMMA). Ends with a
// barrier so C is visible and A/B are safe to overwrite afterwards.
__device__ __forceinline__ void mm64(float* __restrict__ C,
                                     const float* __restrict__ A,
                                     const float* __restrict__ B,
                                     int wid, int lane) {
  const int tm   = wid >> 2;        // tile row 0..3
  const int tn   = wid & 3;         // tile col 0..3
  const int r16  = lane & 15;
  const int half = lane >> 4;
  const int arow = tm * 16 + r16;
  const int bcol = tn * 16 + r16;
  v8f acc = {0.f, 0.f, 0.f, 0.f, 0.f, 0.f, 0.f, 0.f};
#pragma unroll
  for (int kk = 0; kk < 16; ++kk) {   // K = 64 in steps of 4
    const int k0 = kk * 4 + half * 2;
    v2f a, b;
    a.x = A[arow * N + k0];
    a.y = A[arow * N + k0 + 1];
    b.x = B[k0 * N + bcol];
    b.y = B[(k0 + 1) * N + bcol];
    // (neg_a, A, neg_b, B, c_mod, C, reuse_a, reuse_b)
    acc = __builtin_amdgcn_wmma_f32_16x16x4_f32(false, a, false, b,
                                                (short)0, acc, false, false);
  }
#pragma unroll
  for (int v = 0; v < 8; ++v) {
    const int row = tm * 16 + (half ? (v + 8) : v);
    C[row * N + tn * 16 + r16] = acc[v];
  }
  __syncthreads();
}

__global__ __launch_bounds__(NTHREADS, 1)
void spd_log_exp_kernel(const float* __restrict__ f,
                        const float* __restrict__ wts,
                        float* __restrict__ out) {
  extern __shared__ float arena[];     // 4 rotating 64x64 fp32 buffers (64 KB)
  __shared__ float s_red;
  const int b    = blockIdx.x;
  const int tid  = threadIdx.x;
  const int lane = tid & 31;
  const int wid  = tid >> 5;

  float* buf0 = arena;
  float* buf1 = arena + NN;
  float* buf2 = arena + 2 * NN;
  float* buf3 = arena + 3 * NN;

  // ---- load X (16 KB) into LDS, vectorized b128 ----
  {
    const float4* src = (const float4*)(f + (size_t)b * NN);
    float4* dst = (float4*)buf0;
#pragma unroll
    for (int i = 0; i < NN / 4 / NTHREADS; ++i)
      dst[tid + i * NTHREADS] = src[tid + i * NTHREADS];
  }
  if (tid == 0) s_red = 0.f;
  __syncthreads();

  // ---- c = ||X||_F  (>= lambda_max for SPD) ----
  float part = 0.f;
#pragma unroll
  for (int i = 0; i < NN / NTHREADS; ++i) {
    const float v = buf0[tid + i * NTHREADS];
    part += v * v;
  }
  atomicAdd(&s_red, part);             // ds_add_f32
  __syncthreads();
  const float c     = sqrtf(s_red) + 1e-12f;
  const float inv_c = 1.f / c;

  // ---- Y = X / c (spectrum in (0,1]),  Z = I ----
  for (int i = tid; i < NN; i += NTHREADS) {
    buf0[i] *= inv_c;
    const int r = i >> 6, q = i & 63;
    buf1[i] = (r == q) ? 1.f : 0.f;
  }
  __syncthreads();

  float* Y = buf0;
  float* Z = buf1;
  float* T = buf2;
  float* P = buf3;

  // ---- r repeated sqrt via coupled Newton-Schulz:
  //      T = Z*Y; G = 0.5*(3I - T); Y <- Y*G; Z <- G*Z ----
  const int R_LEVELS = 4;
  const int NS_ITERS = 9;
#pragma unroll 1
  for (int lvl = 0; lvl < R_LEVELS; ++lvl) {
#pragma unroll 1
    for (int it = 0; it < NS_ITERS; ++it) {
      mm64(T, Z, Y, wid, lane);                       // T = Z*Y
      for (int i = tid; i < NN; i += NTHREADS) {
        const int r = i >> 6, q = i & 63;
        T[i] = 0.5f * (((r == q) ? 3.f : 0.f) - T[i]); // T := G
      }
      __syncthreads();
      mm64(P, Y, T, wid, lane);                       // newY = Y*G
      mm64(Y, T, Z, wid, lane);                       // newZ -> old Y buffer
      float* nY = P; float* nZ = Y; float* fr = Z;
      Y = nY; Z = nZ; P = fr;                         // rotate buffers
    }
    if (lvl + 1 < R_LEVELS) {                         // reset Z = I
      for (int i = tid; i < NN; i += NTHREADS) {
        const int r = i >> 6, q = i & 63;
        Z[i] = (r == q) ? 1.f : 0.f;
      }
      __syncthreads();
    }
  }

  // ---- D = R - I,  R = (X/c)^(1/2^r) ~ I ----
  float* D = Y;
  for (int i = tid; i < NN; i += NTHREADS) {
    const int r = i >> 6, q = i & 63;
    D[i] -= (r == q) ? 1.f : 0.f;
  }
  __syncthreads();

  // ---- Mercator log via Horner: H = s_n I + D*H,  s_n = (-1)^(n+1)/n ----
  float* G = Z;
  float* H = T;
  const int LOG_TERMS = 12;
  {
    const float sN = ((LOG_TERMS & 1) ? 1.f : -1.f) / (float)LOG_TERMS;
    for (int i = tid; i < NN; i += NTHREADS) {
      const int r = i >> 6, q = i & 63;
      G[i] = (r == q) ? sN : 0.f;
    }
    __syncthreads();
  }
#pragma unroll 1
  for (int n = LOG_TERMS - 1; n >= 1; --n) {
    mm64(H, D, G, wid, lane);
    const float sn = ((n & 1) ? 1.f : -1.f) / (float)n;
    for (int i = tid; i < NN; i += NTHREADS) {
      const int r = i >> 6, q = i & 63;
      H[i] += (r == q) ? sn : 0.f;
    }
    __syncthreads();
    float* t = G; G = H; H = t;
  }
  float* L = H;                       // free buffer
  mm64(L, D, G, wid, lane);           // L = D*G = log(R)

  // ---- M = 0.5 * w_i (Lf_ij + Lf_ji) w_j, Lf = 2^r L + log(c) I ----
  const float two_r = (float)(1 << R_LEVELS);
  const float logc  = logf(c);
  const float* wv   = wts + (size_t)b * N;
  float* M = D;                       // D buffer is dead now
  for (int i = tid; i < NN; i += NTHREADS) {
    const int r = i >> 6, q = i & 63;
    const float lij = two_r * L[i]         + ((r == q) ? logc : 0.f);
    const float lji = two_r * L[q * N + r] + ((r == q) ? logc : 0.f);
    M[i] = 0.5f * wv[r] * (lij + lji) * wv[q];
  }
  __syncthreads();

  // ---- expm: scale by 2^-s, Taylor-Horner, then square s times ----
  const int SQ = 6;
  const float inv_sq = 1.f / (float)(1 << SQ);
  for (int i = tid; i < NN; i += NTHREADS) M[i] *= inv_sq;
  __syncthreads();

  float* E  = M;
  float* Q  = L;                      // Q = I
  float* Rb = G;                      // scratch
  for (int i = tid; i < NN; i += NTHREADS) {
    const int r = i >> 6, q = i & 63;
    Q[i] = (r == q) ? 1.f : 0.f;
  }
  __syncthreads();
  const int EXP_TERMS = 10;
#pragma unroll 1
  for (int k = EXP_TERMS; k >= 1; --k) {
    mm64(Rb, E, Q, wid, lane);        // Rb = E*Q
    const float invk = 1.f / (float)k;
    for (int i = tid; i < NN; i += NTHREADS) {
      const int r = i >> 6, q = i & 63;
      Rb[i] = Rb[i] * invk + ((r == q) ? 1.f : 0.f);
    }
    __syncthreads();
    float* t = Q; Q = Rb; Rb = t;
  }
#pragma unroll 1
  for (int s = 0; s < SQ; ++s) {      // repeated squaring
    mm64(Rb, Q, Q, wid, lane);
    float* t = Q; Q = Rb; Rb = t;
  }

  // ---- store result ----
  {
    float4* dst = (float4*)(out + (size_t)b * NN);
    const float4* src = (const float4*)Q;
#pragma unroll
    for (int i = 0; i < NN / 4 / NTHREADS; ++i)
      dst[tid + i * NTHREADS] = src[tid + i * NTHREADS];
  }
}

extern "C" void kernel_launch(void* const* d_in, const int* in_sizes, int n_in,
                              void* d_out, int out_size, void* d_ws, size_t ws_size,
                              hipStream_t stream) {
  (void)n_in; (void)d_ws; (void)ws_size; (void)out_size;
  const float* f = (const float*)d_in[0];      // (B,1,64,64) f32
  const float* w = (const float*)d_in[1];      // (B,64)      f32
  float* out = (float*)d_out;                  // (B,1,64,64) f32
  const int B = in_sizes[1] / N;               // 8192
  const size_t shmem = 4 * (size_t)NN * sizeof(float);   // 64 KB arena
  spd_log_exp_kernel<<<B, NTHREADS, shmem, stream>>>(f, w, out);
}